// FeatureDiscriminator_49108656063112
// MI455X (gfx1250) — compile-verified
//
#include <hip/hip_runtime.h>

typedef __attribute__((ext_vector_type(2))) float v2f;
typedef __attribute__((ext_vector_type(8))) float v8f;

#define Bn 256
#define Vn 512
#define Fn 256
#define On 2

// ---------------------------------------------------------------------------
// Kernel 1: Y[b,v,o] = sum_f X[b,v,f] * W[f,o]  via v_wmma_f32_16x16x4_f32.
// One wave computes a 16(v) x 16(n, only n<2 valid) tile, K-loop of 64 WMMAs.
// A 16x4 f32 layout: lanes 0-15 -> M=lane, K={0,1} in vgpr0/1;
//                    lanes16-31 -> M=lane-16, K={2,3}.
// B 4x16 f32 layout (mirror): vgpr0: K=0 (lanes0-15)/K=2 (lanes16-31), vgpr1: K=1/K=3.
// B operand is pre-padded in LDS (cols >= O zeroed, stored as K-pairs) so the
// inner loop has NO divergence: 1 ds_load_b64 + 1 global_load_b64 + 1 wmma.
// C 16x16 f32: vgpr r -> (M=r, N=lane) lanes0-15 ; (M=8+r, N=lane-16) lanes16-31.
// ---------------------------------------------------------------------------
__global__ __launch_bounds__(128) void gemm_xw_wmma(
    const float* __restrict__ X, const float* __restrict__ W, float* __restrict__ Y)
{
    // s_wp[kp*16 + n] = { W[2*kp, n], W[2*kp+1, n] } with zeros for n >= O.
    __shared__ v2f s_wp[(Fn / 2) * 16];          // 128*16 pairs = 16 KB
    int t = threadIdx.x;
    for (int i = t; i < (Fn / 2) * 16; i += 128) {
        int kp = i >> 4;
        int n  = i & 15;
        v2f val;
        val.x = (n < On) ? W[(2 * kp + 0) * On + n] : 0.0f;
        val.y = (n < On) ? W[(2 * kp + 1) * On + n] : 0.0f;
        s_wp[i] = val;
    }
    __syncthreads();

    int lane   = t & 31;
    int waveId = t >> 5;
    int tile   = blockIdx.x * 4 + waveId;        // global 16-row tile index
    int b      = tile >> 5;                      // V/16 = 32 tiles per batch
    int v0     = (tile & 31) << 4;

    int arow  = lane & 15;                       // M within tile
    int akoff = (lane >> 4) << 1;                // 0 or 2 (K sub-offset for A)
    int n     = lane & 15;                       // N for B/C
    int kprel = lane >> 4;                       // 0 or 1 (pair sub-offset for B)

    const float* Ap = X + (((size_t)b * Vn + v0 + arow) * Fn) + akoff;
    const v2f*   Bp = s_wp + kprel * 16 + n;     // advance by 32 pairs per k-step

    v8f c = {};
#pragma unroll 4
    for (int kk = 0; kk < Fn; kk += 4) {
        v2f a  = *(const v2f*)(Ap + kk);         // global_load_b64 (A stream)
        v2f bv = Bp[(kk >> 1) * 16];             // ds_load_b64 (padded B)
        c = __builtin_amdgcn_wmma_f32_16x16x4_f32(false, a, false, bv,
                                                  (short)0, c, false, false);
    }

    int mbase = (lane >> 4) << 3;                // 0 or 8
    if (n < On) {
        float* yp = Y + ((size_t)b * Vn + v0 + mbase) * On + n;
#pragma unroll
        for (int r = 0; r < 8; ++r) yp[(size_t)r * On] = c[r];
    }
}

// ---------------------------------------------------------------------------
// Kernel 2: one block per batch. Fused: degree/dinv, yd = dinv*Y into LDS,
// aggregation z = dinv_v * sum_w a_hat * yd_w + bias, ReLU, dot with head
// weights, block reduce, sigmoid. graphs[b] (1MB) streamed twice; the second
// pass hits L2 (192 MB).
// ---------------------------------------------------------------------------
__global__ __launch_bounds__(512) void gcn_agg_head(
    const int* __restrict__ graphs, const float* __restrict__ Y,
    const float* __restrict__ conv_bias, const float* __restrict__ lin_weight,
    const float* __restrict__ lin_bias, float* __restrict__ out)
{
    __shared__ float s_dinv[Vn];
    __shared__ float s_yd0[Vn];
    __shared__ float s_yd1[Vn];
    __shared__ float s_red[16];

    int b      = blockIdx.x;
    int t      = threadIdx.x;
    int lane   = t & 31;
    int waveId = t >> 5;

    const int* G = graphs + (size_t)b * Vn * Vn;

    // ---- phase 0: degree with forced self-loop; dinv = rsqrt(deg) ----
    for (int v = waveId; v < Vn; v += 16) {
        const int4* grow = (const int4*)(G + (size_t)v * Vn);
        int cnt = 0;
#pragma unroll
        for (int it = 0; it < Vn / 128; ++it) {
            int4 g = grow[it * 32 + lane];
            int wc = it * 128 + lane * 4;
            cnt += (g.x != 0 || wc + 0 == v) ? 1 : 0;
            cnt += (g.y != 0 || wc + 1 == v) ? 1 : 0;
            cnt += (g.z != 0 || wc + 2 == v) ? 1 : 0;
            cnt += (g.w != 0 || wc + 3 == v) ? 1 : 0;
        }
#pragma unroll
        for (int off = 16; off > 0; off >>= 1) cnt += __shfl_xor(cnt, off, 32);
        if (lane == 0) s_dinv[v] = rsqrtf((float)cnt);   // deg >= 1 always
    }
    __syncthreads();

    // ---- phase 1: yd[w] = dinv[w] * Y[b,w,:] into LDS ----
    {
        float d = s_dinv[t];
        s_yd0[t] = d * Y[((size_t)b * Vn + t) * On + 0];
        s_yd1[t] = d * Y[((size_t)b * Vn + t) * On + 1];
    }
    __syncthreads();

    float cb0 = conv_bias[0], cb1 = conv_bias[1];
    float wacc = 0.0f;

    // ---- phase 2: aggregate + ReLU + head dot ----
    for (int v = waveId; v < Vn; v += 16) {
        const int4* grow = (const int4*)(G + (size_t)v * Vn);
        float a0 = 0.0f, a1 = 0.0f;
#pragma unroll
        for (int it = 0; it < Vn / 128; ++it) {
            int4 g = grow[it * 32 + lane];
            int wc = it * 128 + lane * 4;
            if (g.x != 0 || wc + 0 == v) { a0 += s_yd0[wc + 0]; a1 += s_yd1[wc + 0]; }
            if (g.y != 0 || wc + 1 == v) { a0 += s_yd0[wc + 1]; a1 += s_yd1[wc + 1]; }
            if (g.z != 0 || wc + 2 == v) { a0 += s_yd0[wc + 2]; a1 += s_yd1[wc + 2]; }
            if (g.w != 0 || wc + 3 == v) { a0 += s_yd0[wc + 3]; a1 += s_yd1[wc + 3]; }
        }
#pragma unroll
        for (int off = 16; off > 0; off >>= 1) {
            a0 += __shfl_xor(a0, off, 32);
            a1 += __shfl_xor(a1, off, 32);
        }
        float d  = s_dinv[v];
        float r0 = fmaxf(d * a0 + cb0, 0.0f);
        float r1 = fmaxf(d * a1 + cb1, 0.0f);
        wacc += r0 * lin_weight[v * On + 0] + r1 * lin_weight[v * On + 1];
    }

    if (lane == 0) s_red[waveId] = wacc;   // wacc uniform across the wave
    __syncthreads();
    if (t == 0) {
        float s = lin_bias[0];
#pragma unroll
        for (int i = 0; i < 16; ++i) s += s_red[i];
        out[b] = 1.0f / (1.0f + expf(-s));
    }
}

extern "C" void kernel_launch(void* const* d_in, const int* in_sizes, int n_in,
                              void* d_out, int out_size, void* d_ws, size_t ws_size,
                              hipStream_t stream) {
    const float* features    = (const float*)d_in[0];   // [B,V,F] f32
    const int*   graphs      = (const int*)d_in[1];     // [B,V,V] i32
    const float* conv_weight = (const float*)d_in[2];   // [F,O]
    const float* conv_bias   = (const float*)d_in[3];   // [O]
    const float* lin_weight  = (const float*)d_in[4];   // [1, O*V]
    const float* lin_bias    = (const float*)d_in[5];   // [1]
    float* out = (float*)d_out;                         // [B]
    float* Y   = (float*)d_ws;                          // [B,V,O] = 1 MB scratch

    // 8192 waves of 16-row tiles, 4 waves per block
    gemm_xw_wmma<<<(Bn * (Vn / 16)) / 4, 128, 0, stream>>>(features, conv_weight, Y);
    gcn_agg_head<<<Bn, 512, 0, stream>>>(graphs, Y, conv_bias, lin_weight, lin_bias, out);
}